// FeatureWeightNet_78383153152427
// MI455X (gfx1250) — compile-verified
//
#include <hip/hip_runtime.h>
#include <cmath>

// Problem constants (from reference setup_inputs)
constexpr int  Bn = 2, Cn = 64, Hn = 512, Wn = 640;
constexpr long HWn = (long)Hn * Wn;

typedef __attribute__((ext_vector_type(2))) float v2f;
typedef __attribute__((ext_vector_type(8))) float v8f;

__device__ __forceinline__ int refl(int i, int n) {
  i = i < 0 ? -i : i;                    // jnp 'reflect': -k -> k
  return (i >= n) ? (2 * n - 2 - i) : i; // n+k -> n-2-k
}

__device__ __forceinline__ float2 ld2(const float* __restrict__ row, int wc, bool interior) {
  if (interior) return *(const float2*)(row + wc);
  float2 r;
  r.x = row[refl(wc,     Wn)];
  r.y = row[refl(wc + 1, Wn)];
  return r;
}

// ---------------------------------------------------------------------------
// Kernel 0: fold BN into 1x1 conv weights.  Layout in d_ws (floats):
//   [0..127]   W0' (16x8)               [128..143] b0'
//   [144..399] W1pad (16x16, rows>=8=0) [400..407] b1'
//   [408..415] ws                       [416]      bs
// W1 is zero-padded to 16 rows so the MLP kernel's A-operand build needs no
// predication (keeps EXEC untouched around the WMMA sequence).
// ---------------------------------------------------------------------------
__global__ void fw_fold(const float* __restrict__ w0, const float* __restrict__ b0,
                        const float* __restrict__ g0, const float* __restrict__ be0,
                        const float* __restrict__ m0, const float* __restrict__ v0,
                        const float* __restrict__ w1, const float* __restrict__ b1,
                        const float* __restrict__ g1, const float* __restrict__ be1,
                        const float* __restrict__ m1, const float* __restrict__ v1,
                        const float* __restrict__ wsim, const float* __restrict__ bsim,
                        float* __restrict__ wts) {
  const int t = threadIdx.x;
  if (t < 16) {
    float s = g0[t] / sqrtf(v0[t] + 1e-5f);
    for (int i = 0; i < 8; i++) wts[t * 8 + i] = w0[t * 8 + i] * s;
    wts[128 + t] = (b0[t] - m0[t]) * s + be0[t];
    if (t < 8) wts[408 + t] = wsim[t];
    if (t == 0) wts[416] = bsim[0];
  } else {
    int r = t - 16;                       // 0..15: rows of W1pad
    if (r < 8) {
      float s = g1[r] / sqrtf(v1[r] + 1e-5f);
      for (int i = 0; i < 16; i++) wts[144 + r * 16 + i] = w1[r * 16 + i] * s;
      wts[400 + r] = (b1[r] - m1[r]) * s + be1[r];
    } else {
      for (int i = 0; i < 16; i++) wts[144 + r * 16 + i] = 0.f;
    }
  }
}

// ---------------------------------------------------------------------------
// Kernel 1: reflected dual-stencil accumulate + per-group mean.
// Block = one image row (320 threads x 2 pixels). Writes acc to d_out region
// and group means x[pix][8] into d_ws.
// ---------------------------------------------------------------------------
__global__ __launch_bounds__(320) void fw_stencil(const float* __restrict__ ref,
                                                  const float* __restrict__ off,
                                                  float* __restrict__ acc,
                                                  float* __restrict__ xbuf) {
  const int t  = threadIdx.x;
  const int w0 = t * 2;
  const int h  = blockIdx.y;
  const int b  = blockIdx.z;

  // reflected row bases (wide = +-4, narrow = +-2)
  int rB[3] = { refl(h - 4, Hn) * Wn, h * Wn, refl(h + 4, Hn) * Wn };
  int rS[3] = { refl(h - 2, Hn) * Wn, h * Wn, refl(h + 2, Hn) * Wn };
  int cB[3] = { w0 - 4, w0, w0 + 4 };
  int cS[3] = { w0 - 2, w0, w0 + 2 };
  const bool interior = (w0 >= 4) && (w0 + 5 < Wn);

  // 18 offsets for the pixel pair, kept in registers across the channel loop
  const float* ob = off + (long)b * 18 * HWn + (long)h * Wn + w0;
  float2 oB[9], oS[9];
#pragma unroll
  for (int s = 0; s < 9; s++) {
    oB[s] = *(const float2*)(ob + (long)s * HWn);
    oS[s] = *(const float2*)(ob + (long)(s + 9) * HWn);
  }

  const float* rbase = ref + (long)b * Cn * HWn;
  float* ap = acc + (long)b * Cn * HWn + (long)h * Wn + w0;
  const long pix = ((long)b * Hn + h) * (long)Wn + w0;
  float* xp = xbuf + pix * 8;

#pragma unroll 1
  for (int g = 0; g < 8; g++) {          // 8 groups of 8 channels
    float gs0 = 0.f, gs1 = 0.f;
#pragma unroll 1
    for (int cc = 0; cc < 8; cc++) {
      const int c = g * 8 + cc;
      const float* rc = rbase + (long)c * HWn;
      if (b == 0 || c < Cn - 1)          // speculative prefetch of next channel's row
        __builtin_prefetch(rc + HWn + rB[1] + w0, 0, 0);
      float a0 = 0.f, a1 = 0.f;
#pragma unroll
      for (int i = 0; i < 3; i++) {
        const float* rb = rc + rB[i];
        const float* rs = rc + rS[i];
#pragma unroll
        for (int j = 0; j < 3; j++) {
          const int s = i * 3 + j;
          float2 vb = ld2(rb, cB[j], interior);
          float2 vs = ld2(rs, cS[j], interior);
          a0 = fmaf(oB[s].x, vb.x, fmaf(oS[s].x, vs.x, a0));
          a1 = fmaf(oB[s].y, vb.y, fmaf(oS[s].y, vs.y, a1));
        }
      }
      a0 *= 0.5f; a1 *= 0.5f;
      *(float2*)(ap + (long)c * HWn) = make_float2(a0, a1);
      gs0 += a0; gs1 += a1;
    }
    xp[g]     = gs0 * 0.125f;            // group mean (C/G == 8)
    xp[8 + g] = gs1 * 0.125f;
  }
}

// ---------------------------------------------------------------------------
// Kernel 2: per-pixel MLP via V_WMMA_F32_16X16X4_F32.
// Wave handles 16 pixels (N dim). conv0: 16x8 * 8x16 (2 K-chunks),
// conv1: 16(zero-padded)x16 * 16x16 (4 K-chunks), then 8->1 dot + sigmoid.
// ---------------------------------------------------------------------------
__global__ __launch_bounds__(128) void fw_mlp(const float* __restrict__ xbuf,
                                              const float* __restrict__ wg,
                                              float* __restrict__ out) {
  __shared__ float wts[448];
  __shared__ float ldsH[4][16][16];
  const int tid = threadIdx.x;
  for (int i = tid; i < 417; i += 128) wts[i] = wg[i];
  __syncthreads();

  const int  wv    = tid >> 5;
  const int  lane  = tid & 31;
  const int  n     = lane & 15;              // A-row / B-col / D-col for this lane
  const int  khalf = (lane < 16) ? 0 : 2;    // K-pair selected by lane half
  const int  mbias = (lane < 16) ? 0 : 8;    // D-row half
  const long p0    = ((long)blockIdx.x * 4 + wv) * 16;

  const float* W0p = wts;            // 16x8
  const float* b0p = wts + 128;
  const float* W1p = wts + 144;      // 16x16 (rows >= 8 are zero)
  const float* b1p = wts + 400;
  const float* wsp = wts + 408;
  const float  bsv = wts[416];

  const float* xp = xbuf + (p0 + n) * 8;

  // ---- conv0: D[16x16] = W0'[16x8] x X[8x16] ----
  v8f c0 = {0.f,0.f,0.f,0.f,0.f,0.f,0.f,0.f};
#pragma unroll
  for (int k0 = 0; k0 < 8; k0 += 4) {
    const int kk = k0 + khalf;
    v2f a, bm;
    a[0]  = W0p[n * 8 + kk];
    a[1]  = W0p[n * 8 + kk + 1];
    bm[0] = xp[kk];
    bm[1] = xp[kk + 1];
    c0 = __builtin_amdgcn_wmma_f32_16x16x4_f32(false, a, false, bm, (short)0, c0, false, false);
  }
  // bias + relu, re-layout M(hidden)->K through LDS
#pragma unroll
  for (int r = 0; r < 8; r++) {
    float hv = c0[r] + b0p[r + mbias];
    ldsH[wv][r + mbias][n] = hv > 0.f ? hv : 0.f;
  }
  __syncthreads();

  // ---- conv1: D[16x16] = W1pad[16x16] x H[16x16] ----
  v8f c1 = {0.f,0.f,0.f,0.f,0.f,0.f,0.f,0.f};
#pragma unroll
  for (int k0 = 0; k0 < 16; k0 += 4) {
    const int kk = k0 + khalf;
    v2f a, bm;
    a[0]  = W1p[n * 16 + kk];          // rows >= 8 are zero in memory: no predication
    a[1]  = W1p[n * 16 + kk + 1];
    bm[0] = ldsH[wv][kk][n];
    bm[1] = ldsH[wv][kk + 1][n];
    c1 = __builtin_amdgcn_wmma_f32_16x16x4_f32(false, a, false, bm, (short)0, c1, false, false);
  }

  // lane n (<16) holds D[M=0..7][N=n] in c1[0..7]: finish 8->1 + sigmoid
  if (lane < 16) {
    float sim = bsv;
#pragma unroll
    for (int r = 0; r < 8; r++) {
      float hv = c1[r] + b1p[r];
      hv = hv > 0.f ? hv : 0.f;
      sim = fmaf(wsp[r], hv, sim);
    }
    out[p0 + lane] = 1.f / (1.f + __expf(-sim));
  }
}

// ---------------------------------------------------------------------------
extern "C" void kernel_launch(void* const* d_in, const int* in_sizes, int n_in,
                              void* d_out, int out_size, void* d_ws, size_t ws_size,
                              hipStream_t stream) {
  (void)in_sizes; (void)n_in; (void)out_size; (void)ws_size;
  const float* ref  = (const float*)d_in[0];
  const float* off  = (const float*)d_in[1];
  const float* w0   = (const float*)d_in[2];
  const float* b0   = (const float*)d_in[3];
  const float* g0   = (const float*)d_in[4];
  const float* be0  = (const float*)d_in[5];
  const float* m0   = (const float*)d_in[6];
  const float* v0   = (const float*)d_in[7];
  const float* w1   = (const float*)d_in[8];
  const float* b1   = (const float*)d_in[9];
  const float* g1   = (const float*)d_in[10];
  const float* be1  = (const float*)d_in[11];
  const float* m1   = (const float*)d_in[12];
  const float* v1   = (const float*)d_in[13];
  const float* wsim = (const float*)d_in[14];
  const float* bsim = (const float*)d_in[15];

  float* out_sig = (float*)d_out;                      // [B,H,W] sigmoid
  float* out_acc = out_sig + (long)Bn * Hn * Wn;       // [B,C,H,W] acc
  float* wts  = (float*)d_ws;                          // folded weights (417 floats)
  float* xbuf = wts + 448;                             // [B*H*W][8] group means

  hipLaunchKernelGGL(fw_fold, dim3(1), dim3(32), 0, stream,
                     w0, b0, g0, be0, m0, v0, w1, b1, g1, be1, m1, v1, wsim, bsim, wts);
  hipLaunchKernelGGL(fw_stencil, dim3(1, Hn, Bn), dim3(320), 0, stream,
                     ref, off, out_acc, xbuf);
  const int nblk = (Bn * Hn * Wn) / 64;                // 4 waves x 16 px per block
  hipLaunchKernelGGL(fw_mlp, dim3(nblk), dim3(128), 0, stream,
                     xbuf, wts, out_sig);
}